// IPS_16587163697209
// MI455X (gfx1250) — compile-verified
//
#include <hip/hip_runtime.h>
#include <cstdint>

#define T_SAMP 5
#define NPLANE 256          // b*c = 8*32
#define HW     65536        // 256*256
#define NSLICE 1280         // T*NPLANE
#define ATT_N  16777216     // 8*32*256*256

#define USE_ASYNC_LDS 1

typedef __attribute__((ext_vector_type(2))) float v2f;
typedef __attribute__((ext_vector_type(8))) float v8f;

// ---------- helpers ----------
__device__ __forceinline__ float sigmoidf_(float v) {
    // numerically-stable sigmoid (matches jax.nn.sigmoid behavior)
    float e   = __expf(-fabsf(v));
    float num = (v >= 0.f) ? 1.f : e;
    return __fdividef(num, 1.f + e);
}

__device__ __forceinline__ void gauss7(float* k1) {
    float s = 0.f;
#pragma unroll
    for (int d = 0; d < 7; ++d) {
        float t = (float)(d - 3) / 1.5f;      // sigma = 1.5
        float e = __expf(-0.5f * t * t);
        k1[d] = e; s += e;
    }
#pragma unroll
    for (int d = 0; d < 7; ++d) k1[d] /= s;
}

// 1-D blurred-delta tap with reflect padding folded in.
// r0 = delta row, o in [0,7) indexes output row (r0-3+o).
__device__ __forceinline__ float krAt(int r0, int o, const float* k1) {
    int r = r0 - 3 + o;
    if (r < 0 || r > 255) return 0.f;
    float acc = 0.f;
#pragma unroll
    for (int dr = 0; dr < 7; ++dr) {
        int i  = r + dr - 3;
        int rr = (i < 0) ? -i : ((i > 255) ? 510 - i : i);   // 'reflect' pad
        acc += (rr == r0) ? k1[dr] : 0.f;
    }
    return acc;
}

#if USE_ASYNC_LDS
__device__ __forceinline__ void issue_tile(const float4* us, int tid,
                                           uint32_t ldsbase, int tile) {
    const float4* src = us + tile * 1024;    // 1024 float4 = 4096 floats per tile
#pragma unroll
    for (int j = 0; j < 4; ++j) {
        int o = j * 256 + tid;
        asm volatile("global_load_async_to_lds_b128 %0, %1, off"
                     :: "v"(ldsbase + (uint32_t)(o * 16)),
                        "v"((uint64_t)(uintptr_t)(src + o))
                     : "memory");
    }
}
#endif

// ---------- K1: fused sigmoid(x)*urnd argmax/max over each (t,b,c) slice ----------
__global__ __launch_bounds__(256) void k_reduce(const float* __restrict__ x,
                                                const float* __restrict__ u,
                                                float* __restrict__ maxv,
                                                int* __restrict__ argi)
{
    const int p   = blockIdx.x;            // (t,b,c) slice: t = p>>8, bc = p&255
    const int bc  = p & (NPLANE - 1);
    const int tid = threadIdx.x;
    const float4* __restrict__ xs = (const float4*)(x + (size_t)bc * HW);
    const float4* __restrict__ us = (const float4*)(u + (size_t)p  * HW);

    float best = -__builtin_inff();
    int   bidx = 0;

#if USE_ASYNC_LDS
    __shared__ __align__(16) float tiles[2][4096];   // 32 KB double buffer
    uint32_t lb0 = (uint32_t)(uintptr_t)&tiles[0][0];
    uint32_t lb1 = (uint32_t)(uintptr_t)&tiles[1][0];

    issue_tile(us, tid, lb0, 0);
    for (int tile = 0; tile < 16; ++tile) {
        if (tile < 15) {
            issue_tile(us, tid, ((tile + 1) & 1) ? lb1 : lb0, tile + 1);
            asm volatile("s_wait_asynccnt 0x4" ::: "memory");  // current tile landed
        } else {
            asm volatile("s_wait_asynccnt 0x0" ::: "memory");
        }
        __syncthreads();
        const float4* lbuf = (const float4*)&tiles[tile & 1][0];
#pragma unroll
        for (int j = 0; j < 4; ++j) {
            int    o  = j * 256 + tid;
            float4 uv = lbuf[o];
            int    v4 = tile * 1024 + o;
            float4 xv = xs[v4];
            float c0 = sigmoidf_(xv.x) * uv.x;
            float c1 = sigmoidf_(xv.y) * uv.y;
            float c2 = sigmoidf_(xv.z) * uv.z;
            float c3 = sigmoidf_(xv.w) * uv.w;
            int base = v4 * 4;                       // strict > keeps first occurrence
            if (c0 > best) { best = c0; bidx = base;     }
            if (c1 > best) { best = c1; bidx = base + 1; }
            if (c2 > best) { best = c2; bidx = base + 2; }
            if (c3 > best) { best = c3; bidx = base + 3; }
        }
        __syncthreads();
    }
#else
    for (int i = 0; i < 64; ++i) {
        int    v4 = i * 256 + tid;
        float4 uv = us[v4];
        float4 xv = xs[v4];
        float c0 = sigmoidf_(xv.x) * uv.x;
        float c1 = sigmoidf_(xv.y) * uv.y;
        float c2 = sigmoidf_(xv.z) * uv.z;
        float c3 = sigmoidf_(xv.w) * uv.w;
        int base = v4 * 4;
        if (c0 > best) { best = c0; bidx = base;     }
        if (c1 > best) { best = c1; bidx = base + 1; }
        if (c2 > best) { best = c2; bidx = base + 2; }
        if (c3 > best) { best = c3; bidx = base + 3; }
    }
#endif

    __shared__ float sv[256];
    __shared__ int   si[256];
    sv[tid] = best; si[tid] = bidx;
    __syncthreads();
    for (int s = 128; s > 0; s >>= 1) {
        if (tid < s) {
            float ov = sv[tid + s]; int oi = si[tid + s];
            float cv = sv[tid];     int ci = si[tid];
            if (ov > cv || (ov == cv && oi < ci)) { sv[tid] = ov; si[tid] = oi; }
        }
        __syncthreads();
    }
    if (tid == 0) { maxv[p] = sv[0]; argi[p] = si[0]; }
}

// ---------- K2: joints stats, visibility, stamp location, global max ----------
__global__ __launch_bounds__(256) void k_stats(const float* __restrict__ maxv,
                                               const int* __restrict__ argi,
                                               float* __restrict__ tail,
                                               int* __restrict__ planeRC,
                                               int* __restrict__ gmaxI)
{
    int i = threadIdx.x;                       // plane = b*32+c
    float ji[T_SAMP], jj[T_SAMP];
    int visn = 0;
#pragma unroll
    for (int t = 0; t < T_SAMP; ++t) {
        float mx = maxv[t * NPLANE + i];
        int   a  = argi[t * NPLANE + i];
        bool  v  = (mx >= 0.1f);
        ji[t] = v ? (float)(a >> 8)  * (1.f / 256.f) : 0.f;
        jj[t] = v ? (float)(a & 255) * (1.f / 256.f) : 0.f;
        visn += v ? 1 : 0;
    }
    float mi = 0.f, mj = 0.f;
#pragma unroll
    for (int t = 0; t < T_SAMP; ++t) { mi += ji[t]; mj += jj[t]; }
    mi /= 5.0f; mj /= 5.0f;
    float vi = 0.f, vj = 0.f;
#pragma unroll
    for (int t = 0; t < T_SAMP; ++t) {
        float di = ji[t] - mi; vi += di * di;
        float dj = jj[t] - mj; vj += dj * dj;
    }
    vi /= 4.0f; vj /= 4.0f;                    // unbiased (ddof=1)
    bool vis = (visn >= 3);                    // mean(vis_t) > 0.5
    if (!vis) { mi = 0.f; mj = 0.f; vi = 0.f; vj = 0.f; }

    tail[i * 2 + 0]        = mi;  tail[i * 2 + 1]        = mj;   // m_joints
    tail[512 + i * 2 + 0]  = vi;  tail[512 + i * 2 + 1]  = vj;   // v_joints
    tail[1024 + i]         = vis ? 1.f : 0.f;                     // vis

    int row = (int)rintf(mi * 256.f); row = row < 0 ? 0 : (row > 255 ? 255 : row);
    int col = (int)rintf(mj * 256.f); col = col < 0 ? 0 : (col > 255 ? 255 : col);
    planeRC[i * 4 + 0] = row;
    planeRC[i * 4 + 1] = col;
    planeRC[i * 4 + 2] = vis ? 1 : 0;
    planeRC[i * 4 + 3] = 0;

    if (vis) {
        float k1[7]; gauss7(k1);
        float mr = 0.f, mc = 0.f;
#pragma unroll
        for (int o = 0; o < 7; ++o) {
            mr = fmaxf(mr, krAt(row, o, k1));
            mc = fmaxf(mc, krAt(col, o, k1));
        }
        atomicMax(gmaxI, __float_as_int(mr * mc));   // values >= 0: int order == float order
    }
}

// ---------- K2b: blurred stamp = Kr (x) Kc via one WMMA rank-1 matmul per plane ----------
__global__ __launch_bounds__(32) void k_stamp(const int* __restrict__ planeRC,
                                              float* __restrict__ patch)
{
    int plane = blockIdx.x;
    int lane  = threadIdx.x;
    int row = planeRC[plane * 4 + 0];
    int col = planeRC[plane * 4 + 1];

    float k1[7]; gauss7(k1);
    float kr = (lane < 7) ? krAt(row, lane, k1) : 0.f;
    float kc = (lane < 7) ? krAt(col, lane, k1) : 0.f;

    // A(16x4): column K=0 = Kr (lanes 0-15 hold M=lane, K=0 in VGPR0 low half)
    // B(4x16): row    K=0 = Kc
    v2f a; a.x = kr; a.y = 0.f;
    v2f b; b.x = kc; b.y = 0.f;
    v8f c = {};
    v8f d = __builtin_amdgcn_wmma_f32_16x16x4_f32(false, a, false, b,
                                                  (short)0, c, false, false);
    // D 16x16 f32 layout: VGPR g -> M=g (lanes 0-15) / M=8+g (lanes 16-31), N=lane&15
    int colI  = lane & 15;
    int rbase = (lane >> 4) * 8;
#pragma unroll
    for (int g = 0; g < 8; ++g) {
        int r = rbase + g;
        if (r < 7 && colI < 7) patch[plane * 49 + r * 7 + colI] = d[g];
    }
}

// ---------- K3: write att = patch/gmax (min==0), zeros elsewhere ----------
__global__ __launch_bounds__(256) void k_att(const int* __restrict__ planeRC,
                                             const float* __restrict__ patch,
                                             const int* __restrict__ gmaxI,
                                             float* __restrict__ att)
{
    int plane = blockIdx.x >> 6;              // 64 blocks per 65536-pixel plane
    int chunk = blockIdx.x & 63;
    int tid   = threadIdx.x;
    int pix   = chunk * 1024 + tid * 4;
    int r = pix >> 8;
    int c = pix & 255;

    int row = planeRC[plane * 4 + 0];
    int col = planeRC[plane * 4 + 1];
    int vis = planeRC[plane * 4 + 2];
    float g     = __int_as_float(*gmaxI);
    float scale = 1.0f / g;                   // g==0 -> inf; 0*inf = NaN == ref 0/0

    int  ro  = r - (row - 3);
    bool rin = (vis != 0) && (ro >= 0) && (ro < 7);
    float vv[4];
#pragma unroll
    for (int k = 0; k < 4; ++k) {
        int   co   = (c + k) - (col - 3);
        float base = (rin && co >= 0 && co < 7) ? patch[plane * 49 + ro * 7 + co] : 0.f;
        vv[k] = base * scale;
    }
    float4 o4; o4.x = vv[0]; o4.y = vv[1]; o4.z = vv[2]; o4.w = vv[3];
    ((float4*)att)[((size_t)plane << 14) + (size_t)(pix >> 2)] = o4;
}

// ---------- launch ----------
extern "C" void kernel_launch(void* const* d_in, const int* in_sizes, int n_in,
                              void* d_out, int out_size, void* d_ws, size_t ws_size,
                              hipStream_t stream) {
    const float* x    = (const float*)d_in[0];   // [8,32,256,256]
    const float* urnd = (const float*)d_in[1];   // [5,8,32,256,256]
    float* att  = (float*)d_out;                 // [8,32,256,256]
    float* tail = att + ATT_N;                   // m_joints(512) | v_joints(512) | vis(256)

    char*  ws      = (char*)d_ws;
    float* maxv    = (float*)(ws);               // 1280 f32
    int*   argi    = (int*)  (ws + 5120);        // 1280 i32
    int*   gmaxI   = (int*)  (ws + 10240);       // 1 i32 (float bits)
    int*   planeRC = (int*)  (ws + 10304);       // 256 * 4 i32
    float* patch   = (float*)(ws + 10304 + 4096);// 256 * 49 f32

    hipMemsetAsync(gmaxI, 0, sizeof(int), stream);
    k_reduce<<<NSLICE, 256, 0, stream>>>(x, urnd, maxv, argi);
    k_stats <<<1,      256, 0, stream>>>(maxv, argi, tail, planeRC, gmaxI);
    k_stamp <<<NPLANE, 32,  0, stream>>>(planeRC, patch);
    k_att   <<<ATT_N / 1024, 256, 0, stream>>>(planeRC, patch, gmaxI, att);
}